// SpatialGraphConvolutionLayer_39427799777465
// MI455X (gfx1250) — compile-verified
//
#include <hip/hip_runtime.h>
#include <hip/hip_bf16.h>

typedef __attribute__((ext_vector_type(2))) float v2f;
typedef __attribute__((ext_vector_type(8))) float v8f;

#define N_NODES 784
#define B_SZ    512
#define D_SZ    64

// ---------------------------------------------------------------------------
// Stage 1: p[j] = softmax_j(-coords[j]·w_spatial)   (i-part and bias cancel)
// One block, 1024 threads, classic LDS tree reductions (max, then sum).
// ---------------------------------------------------------------------------
__global__ __launch_bounds__(1024) void k_softmax_p(const float* __restrict__ coords,
                                                    const float* __restrict__ w_spatial,
                                                    float* __restrict__ p) {
  __shared__ float buf[1024];
  const int j = threadIdx.x;
  const float w0 = w_spatial[0];
  const float w1 = w_spatial[1];

  float logit = -__builtin_inff();
  if (j < N_NODES) {
    const float s = coords[2 * j] * w0 + coords[2 * j + 1] * w1;
    logit = -s;
  }

  buf[j] = logit;
  __syncthreads();
  for (int off = 512; off > 0; off >>= 1) {
    if (j < off) buf[j] = fmaxf(buf[j], buf[j + off]);
    __syncthreads();
  }
  const float m = buf[0];
  __syncthreads();

  const float e = (j < N_NODES) ? __expf(logit - m) : 0.0f;
  buf[j] = e;
  __syncthreads();
  for (int off = 512; off > 0; off >>= 1) {
    if (j < off) buf[j] += buf[j + off];
    __syncthreads();
  }
  const float Z = buf[0];

  if (j < N_NODES) p[j] = e / Z;
}

// ---------------------------------------------------------------------------
// Stage 2: c[b] = sum_j p[j] * inputs[b, j]   via V_WMMA_F32_16X16X4_F32.
// Block = 128 threads = 4 waves; block handles 16 batch rows; split-K across
// the 4 waves (each wave covers k = wave*4 + 16*iter, 49 iters).
// p is folded into the A fragment; B = ones -> K-permutation invariant.
// A layout (ISA 7.12.2, 32-bit A 16x4): lanes 0-15: VGPR0=K0,VGPR1=K1;
// lanes 16-31: VGPR0=K2,VGPR1=K3. D col 0: lane0 holds M=0..7, lane16 M=8..15.
// ---------------------------------------------------------------------------
__global__ __launch_bounds__(128) void k_reduce_c(const float* __restrict__ x,
                                                  const float* __restrict__ p,
                                                  float* __restrict__ c) {
  __shared__ float sp[N_NODES];
  __shared__ float part[4][16];

  const int tid = threadIdx.x;
  for (int k = tid; k < N_NODES; k += 128) sp[k] = p[k];
  __syncthreads();

  const int wave = tid >> 5;
  const int lane = tid & 31;
  const int half = lane >> 4;       // 0: K0/K1 half, 1: K2/K3 half
  const int m    = lane & 15;       // matrix row within tile
  const int b    = blockIdx.x * 16 + m;
  const float* __restrict__ row = x + (size_t)b * N_NODES;

  v2f ones = {1.0f, 1.0f};
  v8f acc0 = {};
  v8f acc1 = {};

  for (int iter = 0; iter < 49; iter += 2) {
    {
      const int kk = wave * 4 + iter * 16 + 2 * half;
      const float2 xv = *(const float2*)(row + kk);
      v2f a;
      a[0] = xv.x * sp[kk];
      a[1] = xv.y * sp[kk + 1];
      acc0 = __builtin_amdgcn_wmma_f32_16x16x4_f32(false, a, false, ones,
                                                   (short)0, acc0, false, false);
    }
    if (iter + 1 < 49) {
      const int kk = wave * 4 + (iter + 1) * 16 + 2 * half;
      const float2 xv = *(const float2*)(row + kk);
      v2f a;
      a[0] = xv.x * sp[kk];
      a[1] = xv.y * sp[kk + 1];
      acc1 = __builtin_amdgcn_wmma_f32_16x16x4_f32(false, a, false, ones,
                                                   (short)0, acc1, false, false);
    }
  }
  const v8f acc = acc0 + acc1;

  // Column 0 of D: lane 0 -> M=0..7, lane 16 -> M=8..15 (8 acc VGPRs).
  if (lane == 0) {
#pragma unroll
    for (int v = 0; v < 8; ++v) part[wave][v] = acc[v];
  }
  if (lane == 16) {
#pragma unroll
    for (int v = 0; v < 8; ++v) part[wave][8 + v] = acc[v];
  }
  __syncthreads();

  if (tid < 16) {
    c[blockIdx.x * 16 + tid] =
        part[0][tid] + part[1][tid] + part[2][tid] + part[3][tid];
  }
}

// ---------------------------------------------------------------------------
// Stage 3: out[b, i, d] = c[b] * w_dense[d].  Pure bandwidth: one float4
// (global_store_b128) per thread, coalesced along d.  grid=(49, 512),
// block=256: 49*256 = 784 rows * 16 float4s per b.
// ---------------------------------------------------------------------------
__global__ __launch_bounds__(256) void k_broadcast_out(const float* __restrict__ c,
                                                       const float* __restrict__ w_dense,
                                                       float* __restrict__ out) {
  const int b  = blockIdx.y;
  const int t  = blockIdx.x * 256 + threadIdx.x;  // 0 .. 12543
  const int d4 = t & 15;
  const int i  = t >> 4;

  const float  cb = c[b];
  const float4 w4 = ((const float4*)w_dense)[d4];
  const float4 val = make_float4(cb * w4.x, cb * w4.y, cb * w4.z, cb * w4.w);

  ((float4*)out)[((size_t)b * N_NODES + i) * (D_SZ / 4) + d4] = val;
}

// ---------------------------------------------------------------------------
extern "C" void kernel_launch(void* const* d_in, const int* in_sizes, int n_in,
                              void* d_out, int out_size, void* d_ws, size_t ws_size,
                              hipStream_t stream) {
  const float* inputs    = (const float*)d_in[0];  // (512, 784)
  const float* coords    = (const float*)d_in[1];  // (784, 2)
  const float* w_spatial = (const float*)d_in[2];  // (2, 1)
  // d_in[3] = b_spatial: cancels inside the softmax, unused.
  const float* w_dense   = (const float*)d_in[4];  // (1, 64)
  float* out = (float*)d_out;

  float* p = (float*)d_ws;   // 784 floats
  float* c = p + 1024;       // 512 floats (kept 4KB-aligned region)

  k_softmax_p<<<1, 1024, 0, stream>>>(coords, w_spatial, p);
  k_reduce_c<<<32, 128, 0, stream>>>(inputs, p, c);
  k_broadcast_out<<<dim3(49, 512), 256, 0, stream>>>(c, w_dense, out);
}